// GLiNERCRFWrapper_64441689309624
// MI455X (gfx1250) — compile-verified
//
#include <hip/hip_runtime.h>

// Problem constants (from reference)
#define BB 64
#define SS 2048
#define DD 768
#define TT 2
#define KSTEPS (DD / 4)     // 192 WMMA k-steps of K=4
#define NTILES ((BB * SS) / 16)  // 8192 tiles of 16 (b,s)-rows

typedef __attribute__((ext_vector_type(2))) float v2f;
typedef __attribute__((ext_vector_type(8))) float v8f;

#define NEGINF (-1.0e30f)

__device__ __forceinline__ float lse2(float a, float b) {
    float m = fmaxf(a, b);
    float d = fminf(a, b) - m;          // <= 0
    return m + log1pf(__expf(d));       // exact when d ~ -1e30 (exp->0)
}

// ---------------------------------------------------------------------------
// K0: build the WMMA B-operand table (shared by all tiles).
// B is the 4x16 (KxN) operand of V_WMMA_F32_16X16X4_F32:
//   lanes 0-15 : v0 = B[K=0][N=lane],    v1 = B[K=1][N=lane]
//   lanes 16-31: v0 = B[K=2][N=lane-16], v1 = B[K=3][N=lane-16]
// B[k][n] = proj_w[n][k_global] for n < T, else 0 (N padded 2 -> 16).
// Layout: bop[ks*64 + lane*2 + {0,1}]
// ---------------------------------------------------------------------------
__global__ __launch_bounds__(256) void k0_build_bop(
    const float* __restrict__ proj_w, float* __restrict__ bop)
{
    int idx = blockIdx.x * blockDim.x + threadIdx.x;   // 0 .. 192*32-1
    if (idx >= KSTEPS * 32) return;
    int ks   = idx >> 5;
    int lane = idx & 31;
    int n    = lane & 15;
    int koff = (lane >> 4) * 2;
    int k    = ks * 4 + koff;
    float f0 = 0.0f, f1 = 0.0f;
    if (n < TT) {
        f0 = proj_w[n * DD + k];
        f1 = proj_w[n * DD + k + 1];
    }
    bop[idx * 2 + 0] = f0;
    bop[idx * 2 + 1] = f1;
}

// ---------------------------------------------------------------------------
// K1: projection GEMM via V_WMMA_F32_16X16X4_F32.
// One wave per tile of 16 consecutive flattened rows r = b*S + s.
// A (16x4, f32): lanes 0-15 hold row M=lane, K=k+0/k+1; lanes 16-31 hold
// row M=lane-16, K=k+2/k+3  -> each lane does one 8B load per k-step.
// D (16x16 f32): lane<16: acc[r] = D[M=r][N=lane]; lane>=16: D[M=r+8][N=lane-16].
// Only N=0,1 are real tags: add bias and store emissions [row][t].
// ---------------------------------------------------------------------------
__global__ __launch_bounds__(256) void k1_gemm_wmma(
    const float* __restrict__ emb,     // B*S*D
    const float* __restrict__ bop,     // KSTEPS*64
    const float* __restrict__ proj_b,  // T
    float* __restrict__ em)            // B*S*T
{
    int lane = threadIdx.x & 31;
    int warp = threadIdx.x >> 5;
    int tile = blockIdx.x * (blockDim.x >> 5) + warp;
    if (tile >= NTILES) return;

    long r0    = (long)tile * 16;
    int  mrow  = lane & 15;
    int  hi    = lane >> 4;                       // 0 or 1
    const float* ap = emb + (r0 + mrow) * (long)DD + hi * 2;
    const float* bp = bop + lane * 2;

    v8f acc = {};
#pragma unroll 4
    for (int ks = 0; ks < KSTEPS; ++ks) {
        v2f a = *(const v2f*)(ap + 4 * ks);       // 8B, 8B-aligned
        v2f b = *(const v2f*)(bp + 64 * ks);      // 8B, coalesced, L2-hot
        acc = __builtin_amdgcn_wmma_f32_16x16x4_f32(
            /*neg_a=*/false, a, /*neg_b=*/false, b,
            /*c_mod=*/(short)0, acc, /*reuse_a=*/false, /*reuse_b=*/false);
    }

    int n = mrow;                                  // N index this lane holds
    if (n < TT) {
        float pb = proj_b[n];
        int mbase = hi * 8;
#pragma unroll
        for (int r = 0; r < 8; ++r)
            em[(r0 + mbase + r) * TT + n] = acc[r] + pb;
    }
}

// ---------------------------------------------------------------------------
// K2: CRF log-likelihood per batch row.
// Forward recurrence == ordered product of 2x2 log-semiring matrices
//   M_s[i][j] = trans[i][j] + em[s][j]   (identity when mask[s]==0).
// 256 threads x 8 steps each, then order-preserving adjacent-segment
// reduction in LDS (log-matmul is non-commutative).
// ---------------------------------------------------------------------------
__global__ __launch_bounds__(256) void k2_crf(
    const float* __restrict__ em,      // B*S*T
    const int*   __restrict__ labels,  // B*S
    const int*   __restrict__ mask,    // B*S
    const float* __restrict__ start,   // T
    const float* __restrict__ endt,    // T
    const float* __restrict__ trans,   // T*T
    float* __restrict__ llh)           // B
{
    __shared__ float4 sP[256];
    __shared__ float  sNum[256];
    __shared__ int    sLen[256];

    int b   = blockIdx.x;
    int tid = threadIdx.x;
    const float* e = em     + (long)b * SS * TT;
    const int*   y = labels + (long)b * SS;
    const int*   m = mask   + (long)b * SS;

    float t00 = trans[0], t01 = trans[1], t10 = trans[2], t11 = trans[3];

    // identity in log semiring
    float p00 = 0.0f, p01 = NEGINF, p10 = NEGINF, p11 = 0.0f;
    float numsum = 0.0f;
    int   lensum = (tid == 0) ? m[0] : 0;

    int s0 = 1 + tid * 8;
#pragma unroll
    for (int i = 0; i < 8; ++i) {
        int s = s0 + i;
        if (s < SS) {
            int  mk = m[s];
            lensum += mk;
            float e0 = e[2 * s], e1 = e[2 * s + 1];
            if (mk) {
                float n00 = lse2(p00 + t00, p01 + t10) + e0;
                float n01 = lse2(p00 + t01, p01 + t11) + e1;
                float n10 = lse2(p10 + t00, p11 + t10) + e0;
                float n11 = lse2(p10 + t01, p11 + t11) + e1;
                p00 = n00; p01 = n01; p10 = n10; p11 = n11;
                int yp = (y[s - 1] > 0) ? 1 : 0;
                int yc = (y[s] > 0) ? 1 : 0;
                numsum += trans[yp * 2 + yc] + (yc ? e1 : e0);
            }
        }
    }

    sP[tid]   = make_float4(p00, p01, p10, p11);
    sNum[tid] = numsum;
    sLen[tid] = lensum;
    __syncthreads();

    // Ordered reduction: combine adjacent segments (left tid covers earlier s)
    for (int st = 1; st < 256; st <<= 1) {
        if ((tid & (2 * st - 1)) == 0) {
            float4 A = sP[tid], Bm = sP[tid + st];
            float4 C;
            C.x = lse2(A.x + Bm.x, A.y + Bm.z);
            C.y = lse2(A.x + Bm.y, A.y + Bm.w);
            C.z = lse2(A.z + Bm.x, A.w + Bm.z);
            C.w = lse2(A.z + Bm.y, A.w + Bm.w);
            sP[tid] = C;
            sNum[tid] += sNum[tid + st];
            sLen[tid] += sLen[tid + st];
        }
        __syncthreads();
    }

    if (tid == 0) {
        float4 P = sP[0];
        float e00 = e[0], e01 = e[1];
        float a0 = start[0] + e00, a1 = start[1] + e01;
        float f0 = lse2(a0 + P.x, a1 + P.z);
        float f1 = lse2(a0 + P.y, a1 + P.w);
        float denom = lse2(f0 + endt[0], f1 + endt[1]);

        int len   = sLen[0];
        int y0    = (y[0] > 0) ? 1 : 0;
        int ylast = (y[len - 1] > 0) ? 1 : 0;
        float num = start[y0] + (y0 ? e01 : e00) + sNum[0] + endt[ylast];
        llh[b] = num - denom;
    }
}

// ---------------------------------------------------------------------------
// K3: loss = ALPHA*base + BETA*(-mean(llh))
// ---------------------------------------------------------------------------
__global__ __launch_bounds__(64) void k3_finalize(
    const float* __restrict__ llh, const float* __restrict__ base,
    float* __restrict__ out)
{
    __shared__ float s[64];
    int tid = threadIdx.x;
    s[tid] = llh[tid];
    __syncthreads();
    for (int off = 32; off >= 1; off >>= 1) {
        if (tid < off) s[tid] += s[tid + off];
        __syncthreads();
    }
    if (tid == 0) {
        float mean = s[0] * (1.0f / (float)BB);
        out[0] = 0.7f * base[0] + 0.3f * (-mean);
    }
}

extern "C" void kernel_launch(void* const* d_in, const int* in_sizes, int n_in,
                              void* d_out, int out_size, void* d_ws, size_t ws_size,
                              hipStream_t stream)
{
    const float* emb    = (const float*)d_in[0]; // token_embeddings B*S*D
    const float* proj_w = (const float*)d_in[1]; // T*D
    const float* proj_b = (const float*)d_in[2]; // T
    const float* start  = (const float*)d_in[3]; // T
    const float* endt   = (const float*)d_in[4]; // T
    const float* trans  = (const float*)d_in[5]; // T*T
    const float* base   = (const float*)d_in[6]; // 1
    const int*   labels = (const int*)d_in[7];   // B*S
    const int*   mask   = (const int*)d_in[8];   // B*S

    float* ws  = (float*)d_ws;
    float* bop = ws;                                  // KSTEPS*64   = 12288 floats
    float* em  = ws + KSTEPS * 64;                    // B*S*T       = 262144 floats
    float* llh = ws + KSTEPS * 64 + BB * SS * TT;     // B           = 64 floats

    k0_build_bop<<<(KSTEPS * 32 + 255) / 256, 256, 0, stream>>>(proj_w, bop);
    k1_gemm_wmma<<<NTILES / 8, 256, 0, stream>>>(emb, bop, proj_b, em);
    k2_crf<<<BB, 256, 0, stream>>>(em, labels, mask, start, endt, trans, llh);
    k3_finalize<<<1, 64, 0, stream>>>(llh, base, (float*)d_out);
}